// GCNModel_38397007626710
// MI455X (gfx1250) — compile-verified
//
#include <hip/hip_runtime.h>

typedef __attribute__((ext_vector_type(2))) float v2f;
typedef __attribute__((ext_vector_type(8))) float v8f;

static inline int cdiv(int a, int b) { return (a + b - 1) / b; }

// ---------------------------------------------------------------------------
// C[M x Ncols] = A[M x K] * B[K x Ncols], fp32, one 16x16 tile per wave.
// Uses V_WMMA_F32_16X16X4_F32 (CDNA5). M must be a multiple of 16, K of 4,
// Ncols of 16. blockDim = (32, 4): 4 waves per block, each a different N tile.
// ---------------------------------------------------------------------------
__global__ void gemm_wmma_f32(const float* __restrict__ A,
                              const float* __restrict__ B,
                              float* __restrict__ C,
                              int M, int K, int Ncols) {
  const int lane = threadIdx.x & 31;
  const int grp  = lane >> 4;        // 0: lanes 0-15, 1: lanes 16-31
  const int l16  = lane & 15;
  const int tileM = blockIdx.x;
  const int tileN = blockIdx.y * blockDim.y + threadIdx.y;
  if (tileN * 16 >= Ncols) return;   // wave-uniform: EXEC stays all-ones

  const float* arow = A + (size_t)(tileM * 16 + l16) * K;   // A row for lane
  const float* bcol = B + (tileN * 16 + l16);               // B col for lane

  v8f acc = {};
  for (int k = 0; k < K; k += 4) {
    // A 16x4 layout: VGPR0 = K {0|2}, VGPR1 = K {1|3} by half-wave
    v2f a, b;
    a.x = arow[k + 2 * grp + 0];
    a.y = arow[k + 2 * grp + 1];
    // B 4x16: row K striped across lanes within a VGPR (mirror of A)
    b.x = bcol[(size_t)(k + 2 * grp + 0) * Ncols];
    b.y = bcol[(size_t)(k + 2 * grp + 1) * Ncols];
    acc = __builtin_amdgcn_wmma_f32_16x16x4_f32(
        /*neg_a=*/false, a, /*neg_b=*/false, b,
        /*c_mod=*/(short)0, acc, /*reuse_a=*/false, /*reuse_b=*/false);
  }

  // C/D layout: VGPR v -> M = v + 8*grp, N = l16
  float* crow = C + (size_t)(tileM * 16 + 8 * grp) * Ncols + tileN * 16 + l16;
#pragma unroll
  for (int v = 0; v < 8; ++v)
    crow[(size_t)v * Ncols] = acc[v];
}

// ---------------------------------------------------------------------------
// Degree / normalization helpers
// ---------------------------------------------------------------------------
__global__ void fill_ones(float* __restrict__ p, int n) {
  int i = blockIdx.x * blockDim.x + threadIdx.x;
  if (i < n) p[i] = 1.0f;                    // self-loop contributes 1
}

__global__ void deg_count(const long long* __restrict__ dst,
                          float* __restrict__ deg, int E) {
  int i = blockIdx.x * blockDim.x + threadIdx.x;
  if (i < E) atomicAdd(&deg[(int)dst[i]], 1.0f);
}

__global__ void rsqrt_inplace(float* __restrict__ p, int n) {
  int i = blockIdx.x * blockDim.x + threadIdx.x;
  if (i < n) p[i] = rsqrtf(p[i]);            // deg >= 1 always
}

// ---------------------------------------------------------------------------
// agg init with the self-loop term (and optional bias):
//   out[n,f] = h[n,f] * dinv[n]^2  (+ bias[f])
// ---------------------------------------------------------------------------
__global__ void selfloop_init(const float* __restrict__ h,
                              const float* __restrict__ dinv,
                              const float* __restrict__ bias,
                              float* __restrict__ out, int N, int D) {
  size_t i = (size_t)blockIdx.x * blockDim.x + threadIdx.x;
  size_t total = (size_t)N * D;
  if (i >= total) return;
  int n = (int)(i / D);
  int f = (int)(i % D);
  float di = dinv[n];
  float v = h[i] * di * di;
  if (bias) v += bias[f];
  out[i] = v;
}

// ---------------------------------------------------------------------------
// Edge scatter-add: one wave per edge. Lane l handles float4 chunk l of the
// feature row (D=128 -> all 32 lanes, D=64 -> lanes 0-15).
// out[dst] += h[src] * dinv[src]*dinv[dst]  via global f32 atomics (L2-resident)
// ---------------------------------------------------------------------------
__global__ void edge_aggregate(const float* __restrict__ h,
                               const float* __restrict__ dinv,
                               const long long* __restrict__ src,
                               const long long* __restrict__ dst,
                               float* __restrict__ out, int E, int D) {
  int gid  = blockIdx.x * blockDim.x + threadIdx.x;
  int wave = gid >> 5;
  int lane = threadIdx.x & 31;
  int nw   = (gridDim.x * blockDim.x) >> 5;
  int F4   = D >> 2;
  for (int e = wave; e < E; e += nw) {
    int s = (int)src[e];
    int d = (int)dst[e];
    float nrm = dinv[s] * dinv[d];
    if (lane < F4) {
      const float4* hs = (const float4*)(h + (size_t)s * D);
      float4 v = hs[lane];
      float* o = out + (size_t)d * D + lane * 4;
      atomicAdd(o + 0, v.x * nrm);
      atomicAdd(o + 1, v.y * nrm);
      atomicAdd(o + 2, v.z * nrm);
      atomicAdd(o + 3, v.w * nrm);
    }
  }
}

// ---------------------------------------------------------------------------
// BatchNorm (train-mode, biased variance): column sums / sums-of-squares
// ---------------------------------------------------------------------------
__global__ void zero_stats(float* __restrict__ sum, float* __restrict__ sumsq,
                           int D) {
  int i = blockIdx.x * blockDim.x + threadIdx.x;
  if (i < D) { sum[i] = 0.0f; sumsq[i] = 0.0f; }
}

__global__ void bn_stats(const float* __restrict__ x,
                         float* __restrict__ sum, float* __restrict__ sumsq,
                         int N, int D) {
  int gt = blockIdx.x * blockDim.x + threadIdx.x;
  int f  = gt % D;
  int r0 = gt / D;
  int rstep = (gridDim.x * blockDim.x) / D;
  float s = 0.0f, ss = 0.0f;
  for (int r = r0; r < N; r += rstep) {
    float v = x[(size_t)r * D + f];
    s += v; ss += v * v;
  }
  atomicAdd(&sum[f], s);
  atomicAdd(&sumsq[f], ss);
}

__global__ void bn_apply_relu(float* __restrict__ x,
                              const float* __restrict__ sum,
                              const float* __restrict__ sumsq,
                              const float* __restrict__ g,
                              const float* __restrict__ be, int N, int D) {
  size_t i = (size_t)blockIdx.x * blockDim.x + threadIdx.x;
  size_t total = (size_t)N * D;
  if (i >= total) return;
  int f = (int)(i % D);
  float invn = 1.0f / (float)N;
  float m = sum[f] * invn;
  float var = sumsq[f] * invn - m * m;
  float r = rsqrtf(var + 1e-5f);
  float y = g[f] * (x[i] - m) * r + be[f];
  x[i] = y > 0.0f ? y : 0.0f;
}

// ---------------------------------------------------------------------------
extern "C" void kernel_launch(void* const* d_in, const int* in_sizes, int n_in,
                              void* d_out, int out_size, void* d_ws, size_t ws_size,
                              hipStream_t stream) {
  (void)n_in; (void)out_size; (void)ws_size;
  const float*     x   = (const float*)d_in[0];
  const long long* ei  = (const long long*)d_in[1];
  const float*     W1  = (const float*)d_in[2];
  const float*     g1  = (const float*)d_in[4];
  const float*     be1 = (const float*)d_in[5];
  const float*     W2  = (const float*)d_in[6];
  const float*     g2  = (const float*)d_in[8];
  const float*     be2 = (const float*)d_in[9];
  const float*     W3  = (const float*)d_in[10];
  const float*     b3  = (const float*)d_in[11];
  // b1/b2 cancel exactly in train-mode BatchNorm (mean removes constant shift)

  const int DH = in_sizes[3];            // 128
  const int DO = in_sizes[11];           // 64
  const int DI = in_sizes[2] / DH;       // 128
  const int N  = in_sizes[0] / DI;       // 50000 (= 3125 * 16)
  const int E  = in_sizes[1] / 2;        // 800000
  const long long* src = ei;
  const long long* dst = ei + E;

  float* bufA  = (float*)d_ws;                 // N x DH
  float* bufB  = bufA + (size_t)N * DH;        // N x DH
  float* dinv  = bufB + (size_t)N * DH;        // N   (deg -> dinv in place)
  float* sum   = dinv + N;                     // DH
  float* sumsq = sum + DH;                     // DH
  float* out   = (float*)d_out;                // N x DO

  const dim3 gb(32, 4);                        // 4 waves/block, 1 tile each
  const int eb = cdiv(E * 32, 256);            // one wave per edge
  const int nEl128 = cdiv(N * DH, 256);
  const int statThreads = 256, statBlocks = DH; // DH*256 threads total

  // ---- symmetric normalization coefficients -------------------------------
  fill_ones   <<<cdiv(N, 256), 256, 0, stream>>>(dinv, N);
  deg_count   <<<cdiv(E, 256), 256, 0, stream>>>(dst, dinv, E);
  rsqrt_inplace<<<cdiv(N, 256), 256, 0, stream>>>(dinv, N);

  // ---- layer 1: h = x @ W1 ; aggregate ; BN+ReLU --------------------------
  gemm_wmma_f32<<<dim3(N / 16, cdiv(DH, 64)), gb, 0, stream>>>(x, W1, bufA, N, DI, DH);
  selfloop_init<<<nEl128, 256, 0, stream>>>(bufA, dinv, nullptr, bufB, N, DH);
  edge_aggregate<<<eb, 256, 0, stream>>>(bufA, dinv, src, dst, bufB, E, DH);
  zero_stats   <<<cdiv(DH, 256), 256, 0, stream>>>(sum, sumsq, DH);
  bn_stats     <<<statBlocks, statThreads, 0, stream>>>(bufB, sum, sumsq, N, DH);
  bn_apply_relu<<<nEl128, 256, 0, stream>>>(bufB, sum, sumsq, g1, be1, N, DH);

  // ---- layer 2: h = h1 @ W2 ; aggregate ; BN+ReLU -------------------------
  gemm_wmma_f32<<<dim3(N / 16, cdiv(DH, 64)), gb, 0, stream>>>(bufB, W2, bufA, N, DH, DH);
  selfloop_init<<<nEl128, 256, 0, stream>>>(bufA, dinv, nullptr, bufB, N, DH);
  edge_aggregate<<<eb, 256, 0, stream>>>(bufA, dinv, src, dst, bufB, E, DH);
  zero_stats   <<<cdiv(DH, 256), 256, 0, stream>>>(sum, sumsq, DH);
  bn_stats     <<<statBlocks, statThreads, 0, stream>>>(bufB, sum, sumsq, N, DH);
  bn_apply_relu<<<nEl128, 256, 0, stream>>>(bufB, sum, sumsq, g2, be2, N, DH);

  // ---- layer 3: out = aggregate(h2 @ W3) + b3 -----------------------------
  gemm_wmma_f32<<<dim3(N / 16, cdiv(DO, 64)), gb, 0, stream>>>(bufB, W3, bufA, N, DH, DO);
  selfloop_init<<<cdiv(N * DO, 256), 256, 0, stream>>>(bufA, dinv, b3, out, N, DO);
  edge_aggregate<<<eb, 256, 0, stream>>>(bufA, dinv, src, dst, out, E, DO);
}